// RadarPillarsLoss_80358838108196
// MI455X (gfx1250) — compile-verified
//
#include <hip/hip_runtime.h>
#include <hip/hip_bf16.h>

// ---- problem constants (from reference setup_inputs) ----
#define ALPHA_C   0.25f
#define BETA_C    0.11f
#define REGW_C    2.0f

constexpr int Bn = 32, Hh = 160, Ww = 160, Aa = 3, Cc = 4;
constexpr int HW = Hh * Ww;                 // 25600
constexpr int Nn = HW * Aa;                 // 76800
constexpr int TOTAL = Bn * Nn;              // 2,457,600  (divisible by 256)
constexpr int BLOCK = 256;                  // 8 wave32 waves

typedef float v2f __attribute__((ext_vector_type(2)));
typedef float v8f __attribute__((ext_vector_type(8)));

// Full-precision wave32 sum using the fp32 WMMA as a reduction engine.
// A: VGPR0 = per-lane value (A[m][0]=v[m], A[m][2]=v[m+16]), VGPR1 = 0.
// B: all ones. => D[m][n] = v[m] + v[m+16] for every n.
// Per lane, c[0..7] hold rows {0..7} (lanes 0-15) or {8..15} (lanes 16-31);
// summing them + one shfl_xor(16) yields the full 32-lane sum in all lanes.
// EXEC must be all ones: callers are full blocks with no divergence here.
__device__ __forceinline__ float wave_sum_f32(float v) {
  v2f a; a.x = v;    a.y = 0.0f;
  v2f b; b.x = 1.0f; b.y = 1.0f;
  v8f c = {};
  c = __builtin_amdgcn_wmma_f32_16x16x4_f32(
      /*neg_a=*/false, a, /*neg_b=*/false, b,
      /*c_mod=*/(short)0, c, /*reuse_a=*/false, /*reuse_b=*/false);
  float s = ((c[0] + c[1]) + (c[2] + c[3])) + ((c[4] + c[5]) + (c[6] + c[7]));
  s += __shfl_xor(s, 16, 32);
  return s;
}

__device__ __forceinline__ float ntl(const float* p) {
  return __builtin_nontemporal_load(p);     // streamed once; keep out of L2's way
}

__global__ void rpl_zero_acc(float* __restrict__ acc) {
  if (threadIdx.x < 3) acc[threadIdx.x] = 0.0f;
}

__global__ __launch_bounds__(BLOCK)
void rpl_main(const float* __restrict__ cls_preds,   // (B, N, 4)
              const float* __restrict__ reg_preds,   // (B, N, 7)
              const float* __restrict__ anchors,     // (B, N, 7) after flatten
              const float* __restrict__ gt_boxes,    // (B, HW, 7)
              const int*   __restrict__ gt_labels,   // (B, HW)
              float* __restrict__ acc) {             // [clsSum, regSum, posCnt]
  const int i   = blockIdx.x * BLOCK + threadIdx.x;  // exact launch: i < TOTAL
  const int b   = i / Nn;
  const int n   = i - b * Nn;
  const int bhw = b * HW + n / Aa;                   // shared by the 3 anchors

  // ---------------- classification / focal ----------------
  const float* cp = cls_preds + (size_t)i * Cc;
  const float x0 = ntl(cp + 0), x1 = ntl(cp + 1), x2 = ntl(cp + 2), x3 = ntl(cp + 3);

  const int  label = gt_labels[bhw];                 // cached: reused across A
  const bool pos   = label > 0;
  const int  tgt   = pos ? label : 0;

  const float mx = fmaxf(fmaxf(x0, x1), fmaxf(x2, x3));
  const float e0 = __expf(x0 - mx), e1 = __expf(x1 - mx);
  const float e2 = __expf(x2 - mx), e3 = __expf(x3 - mx);
  const float et = (tgt == 0) ? e0 : (tgt == 1) ? e1 : (tgt == 2) ? e2 : e3;
  const float pt = et / (((e0 + e1) + (e2 + e3)));
  const float alpha = (tgt == 1) ? ALPHA_C : (1.0f - ALPHA_C);
  const float om = 1.0f - pt;
  const float focal = -alpha * om * om * __logf(pt + 1e-6f);  // GAMMA == 2

  // ---------------- regression / smooth-L1 ----------------
  const float* ap = anchors  + (size_t)i * 7;
  const float* rp = reg_preds + (size_t)i * 7;
  const float* gp = gt_boxes + (size_t)bhw * 7;      // cached: reused across A

  const float a0 = ntl(ap + 0), a1 = ntl(ap + 1), a2 = ntl(ap + 2);
  const float a3 = ntl(ap + 3), a4 = ntl(ap + 4), a5 = ntl(ap + 5);
  const float g0 = gp[0], g1 = gp[1], g2 = gp[2];
  const float g3 = gp[3], g4 = gp[4], g5 = gp[5];

  float t[7];
  t[0] = (g0 - a0) / a5;
  t[1] = (g1 - a1) / a4;
  t[2] = (g2 - a2) / a3;
  t[3] = __logf(g3 / a3);
  t[4] = __logf(g4 / a4);
  t[5] = __logf(g5 / a5);
  t[6] = 0.0f;

  const float posf = pos ? 1.0f : 0.0f;
  float l1sum = 0.0f;
#pragma unroll
  for (int j = 0; j < 7; ++j) {
    const float d = fabsf(ntl(rp + j) - t[j] * posf);  // target masked to 0 when !pos
    const float l = (d < BETA_C) ? (0.5f * d * d / BETA_C) : (d - 0.5f * BETA_C);
    l1sum += l;
  }
  const float regp = l1sum * posf;                    // l1 * mask

  // ---------------- wave reduction (WMMA) + one atomic per wave ----------------
  const float cs = wave_sum_f32(focal);
  const float rs = wave_sum_f32(regp);
  const float ps = wave_sum_f32(posf);
  if ((threadIdx.x & 31) == 0) {
    unsafeAtomicAdd(acc + 0, cs);                     // global_atomic_add_f32
    unsafeAtomicAdd(acc + 1, rs);
    unsafeAtomicAdd(acc + 2, ps);
  }
}

__global__ void rpl_finalize(const float* __restrict__ acc, float* __restrict__ out) {
  if (threadIdx.x == 0) {
    const float cls = acc[0] * (1.0f / (float)TOTAL);
    const float reg = acc[1] / (acc[2] + 1e-6f);
    out[0] = cls + REGW_C * reg;   // total
    out[1] = cls;                  // cls_loss
    out[2] = reg;                  // reg_loss
  }
}

extern "C" void kernel_launch(void* const* d_in, const int* in_sizes, int n_in,
                              void* d_out, int out_size, void* d_ws, size_t ws_size,
                              hipStream_t stream) {
  const float* cls_preds = (const float*)d_in[0];
  const float* reg_preds = (const float*)d_in[1];
  const float* anchors   = (const float*)d_in[2];
  const float* gt_boxes  = (const float*)d_in[3];
  const int*   gt_labels = (const int*)d_in[4];
  float* out = (float*)d_out;
  float* acc = (float*)d_ws;

  rpl_zero_acc<<<1, 32, 0, stream>>>(acc);
  rpl_main<<<TOTAL / BLOCK, BLOCK, 0, stream>>>(cls_preds, reg_preds, anchors,
                                                gt_boxes, gt_labels, acc);
  rpl_finalize<<<1, 32, 0, stream>>>(acc, out);
}